// GINLayer_74079595921458
// MI455X (gfx1250) — compile-verified
//
#include <hip/hip_runtime.h>
#include <math.h>

// GIN layer: N=50000 nodes, E=300000 edges, D=256.
// Edge phase is HBM-bound on edge_attr (307MB); x and agg are L2-resident (51MB each, 192MB L2).
// FFN phase uses V_WMMA_F32_16X16X4_F32 (full fp32 semantics); each wave computes a 16x64
// output strip so one A-fragment feeds 4 WMMAs per k-step (1.25 loads/WMMA vs 2).

#define DD      256
#define NNODES  50000
#define NEDGES  300000
#define BN_EPS  1e-5f

typedef __attribute__((ext_vector_type(2))) float v2f;
typedef __attribute__((ext_vector_type(8))) float v8f;

// ---------------------------------------------------------------- zero agg
__global__ __launch_bounds__(256) void k_zero(float4* __restrict__ p, int n4) {
    int i = blockIdx.x * 256 + threadIdx.x;
    if (i < n4) p[i] = make_float4(0.f, 0.f, 0.f, 0.f);
}

// ------------------------------------------------- edge message + scatter-add
// 64 threads per edge, float4 per thread -> fully coalesced 1KB row reads.
// agg (51MB) lives in L2, so the f32 atomics are L2 atomics.
__global__ __launch_bounds__(256) void k_edge_scatter(
    const float* __restrict__ x,
    const long long* __restrict__ ei,      // [2, E] int64
    const float* __restrict__ ea,          // [E, D]
    float* __restrict__ agg)               // [N, D]
{
    int tid = blockIdx.x * 256 + threadIdx.x;
    int e = tid >> 6;                      // edge id
    if (e >= NEDGES) return;
    int j = (tid & 63) * 4;                // feature offset
    int src = (int)ei[e];
    int dst = (int)ei[NEDGES + e];
    float4 xv = *(const float4*)(x  + (size_t)src * DD + j);
    float4 av = *(const float4*)(ea + (size_t)e   * DD + j);
    float m0 = fmaxf(xv.x + av.x, 0.f);
    float m1 = fmaxf(xv.y + av.y, 0.f);
    float m2 = fmaxf(xv.z + av.z, 0.f);
    float m3 = fmaxf(xv.w + av.w, 0.f);
    float* a = agg + (size_t)dst * DD + j;
    atomicAdd(a + 0, m0);
    atomicAdd(a + 1, m1);
    atomicAdd(a + 2, m2);
    atomicAdd(a + 3, m3);
}

// -------------------------------------- h = 2x + agg ; fin = BN1(h)  (fused)
__global__ __launch_bounds__(256) void k_h_bn1(
    const float* __restrict__ x, const float* __restrict__ agg,
    const float* __restrict__ g1, const float* __restrict__ be1,
    const float* __restrict__ mu1, const float* __restrict__ vr1,
    float* __restrict__ h, float* __restrict__ fin)
{
    int i = blockIdx.x * 256 + threadIdx.x;     // one float4 per thread
    if (i >= NNODES * DD / 4) return;
    int c0 = (i & (DD/4 - 1)) * 4;              // feature column of .x
    float4 xv = ((const float4*)x)[i];
    float4 av = ((const float4*)agg)[i];
    float4 hv;
    hv.x = 2.f * xv.x + av.x;
    hv.y = 2.f * xv.y + av.y;
    hv.z = 2.f * xv.z + av.z;
    hv.w = 2.f * xv.w + av.w;
    ((float4*)h)[i] = hv;
    float4 fv;
    { float s = rsqrtf(vr1[c0+0] + BN_EPS) * g1[c0+0]; fv.x = (hv.x - mu1[c0+0]) * s + be1[c0+0]; }
    { float s = rsqrtf(vr1[c0+1] + BN_EPS) * g1[c0+1]; fv.y = (hv.y - mu1[c0+1]) * s + be1[c0+1]; }
    { float s = rsqrtf(vr1[c0+2] + BN_EPS) * g1[c0+2]; fv.z = (hv.z - mu1[c0+2]) * s + be1[c0+2]; }
    { float s = rsqrtf(vr1[c0+3] + BN_EPS) * g1[c0+3]; fv.w = (hv.w - mu1[c0+3]) * s + be1[c0+3]; }
    ((float4*)fin)[i] = fv;
}

// ------------------------------------------------------- WMMA fp32 GEMM strip
// out = A[50000,256] @ W[256,256]^T.  Each wave: one 16(M) x 64(N) strip =
// 4 accumulator tiles; per k-step one A-frag is reused by 4 WMMAs.
// Frag layouts (v_wmma_f32_16x16x4_f32):
//   A: lane holds float2 of row m0+(lane&15) at k + 2*(lane>>4).
//   B (=W^T): float2 of W row n0+(lane&15) at k + 2*(lane>>4)  (row-major W).
//   D: VGPR r -> row m0 + r + 8*(lane>>4), col n0 + (lane&15).
// mode 0: gelu(acc+bias); mode 1: BN2(h + acc + bias).
__global__ __launch_bounds__(256) void k_gemm_wmma(
    const float* __restrict__ A,
    const float* __restrict__ W,
    const float* __restrict__ bias,
    const float* __restrict__ h,           // mode 1 only
    const float* __restrict__ g2, const float* __restrict__ be2,
    const float* __restrict__ mu2, const float* __restrict__ vr2,
    float* __restrict__ out,
    int mode)
{
    const int wave = threadIdx.x >> 5;
    const int lane = threadIdx.x & 31;
    const int task = blockIdx.x * 8 + wave;        // 3125 mtiles * 4 ngroups = 12500
    if (task >= (NNODES / 16) * 4) return;         // wave-uniform guard
    const int m0 = (task >> 2) * 16;
    const int n0 = (task & 3) * 64;

    const int lrow  = lane & 15;
    const int khalf = (lane >> 4) << 1;            // 0 or 2

    const float* arow = A + (size_t)(m0 + lrow) * DD + khalf;
    const float* brow = W + (size_t)(n0 + lrow) * DD + khalf;   // +t*16 rows per tile

    v8f c0 = {}, c1 = {}, c2 = {}, c3 = {};
#pragma unroll 4
    for (int k = 0; k < DD; k += 4) {
        v2f a  = *(const v2f*)(arow + k);
        v2f b0 = *(const v2f*)(brow + k);
        v2f b1 = *(const v2f*)(brow + 16 * DD + k);
        v2f b2 = *(const v2f*)(brow + 32 * DD + k);
        v2f b3 = *(const v2f*)(brow + 48 * DD + k);
        c0 = __builtin_amdgcn_wmma_f32_16x16x4_f32(false, a, false, b0, (short)0, c0, false, false);
        c1 = __builtin_amdgcn_wmma_f32_16x16x4_f32(false, a, false, b1, (short)0, c1, false, false);
        c2 = __builtin_amdgcn_wmma_f32_16x16x4_f32(false, a, false, b2, (short)0, c2, false, false);
        c3 = __builtin_amdgcn_wmma_f32_16x16x4_f32(false, a, false, b3, (short)0, c3, false, false);
    }

    const int row_base = m0 + ((lane >> 4) << 3);
    v8f acc[4] = { c0, c1, c2, c3 };

    if (mode == 0) {
#pragma unroll
        for (int t = 0; t < 4; ++t) {
            const int col = n0 + t * 16 + lrow;
            const float bb = bias[col];
#pragma unroll
            for (int r = 0; r < 8; ++r) {
                float v = acc[t][r] + bb;
                v = 0.5f * v * (1.0f + erff(v * 0.70710678f));   // exact GELU
                out[(size_t)(row_base + r) * DD + col] = v;
            }
        }
    } else {
#pragma unroll
        for (int t = 0; t < 4; ++t) {
            const int col = n0 + t * 16 + lrow;
            const float bb = bias[col];
            const float s  = rsqrtf(vr2[col] + BN_EPS) * g2[col];
            const float sh = be2[col] - mu2[col] * s;
#pragma unroll
            for (int r = 0; r < 8; ++r) {
                size_t idx = (size_t)(row_base + r) * DD + col;
                float z = h[idx] + acc[t][r] + bb;
                out[idx] = z * s + sh;
            }
        }
    }
}

// ---------------------------------------------------------------------------
extern "C" void kernel_launch(void* const* d_in, const int* in_sizes, int n_in,
                              void* d_out, int out_size, void* d_ws, size_t ws_size,
                              hipStream_t stream) {
    const float*     x   = (const float*)d_in[0];
    const long long* ei  = (const long long*)d_in[1];
    const float*     ea  = (const float*)d_in[2];
    const float*     W1  = (const float*)d_in[3];
    const float*     b1  = (const float*)d_in[4];
    const float*     W2  = (const float*)d_in[5];
    const float*     b2  = (const float*)d_in[6];
    const float*     g1  = (const float*)d_in[7];
    const float*     be1 = (const float*)d_in[8];
    const float*     mu1 = (const float*)d_in[9];
    const float*     vr1 = (const float*)d_in[10];
    const float*     g2  = (const float*)d_in[11];
    const float*     be2 = (const float*)d_in[12];
    const float*     mu2 = (const float*)d_in[13];
    const float*     vr2 = (const float*)d_in[14];
    float* out = (float*)d_out;

    const size_t nd = (size_t)NNODES * DD;
    float* agg  = (float*)d_ws;      // region 0: agg, later reused for GELU output
    float* hbuf = agg + nd;          // region 1: h
    float* fin  = hbuf + nd;         // region 2: BN1(h)   (needs 3*51.2MB of ws)

    const int nd4 = (int)(nd / 4);
    const int blocksElem = (nd4 + 255) / 256;            // 12500
    const int blocksEdge = (NEDGES * 64 + 255) / 256;    // 75000
    const int tasks      = (NNODES / 16) * 4;            // 12500 wave strips
    const int blocksGemm = (tasks + 7) / 8;              // 1563

    k_zero<<<blocksElem, 256, 0, stream>>>((float4*)agg, nd4);
    k_edge_scatter<<<blocksEdge, 256, 0, stream>>>(x, ei, ea, agg);
    k_h_bn1<<<blocksElem, 256, 0, stream>>>(x, agg, g1, be1, mu1, vr1, hbuf, fin);
    // GEMM1: gelu(fin @ W1^T + b1) -> reuse agg region
    k_gemm_wmma<<<blocksGemm, 256, 0, stream>>>(fin, W1, b1, nullptr,
                                                nullptr, nullptr, nullptr, nullptr,
                                                agg, 0);
    // GEMM2 + residual + BN2 -> d_out
    k_gemm_wmma<<<blocksGemm, 256, 0, stream>>>(agg, W2, b2, hbuf,
                                                g2, be2, mu2, vr2,
                                                out, 1);
}